// PointPillarsScatter_65128884076644
// MI455X (gfx1250) — compile-verified
//
#include <hip/hip_runtime.h>
#include <hip/hip_bf16.h>
#include <stdint.h>

// Canvas geometry (fixed by the reference): (B=4, C=64, NY=512, NX=512)
#define NXD   512
#define NYD   512
#define NYX   (NYD * NXD)     // 262144 elements per (b,c) plane
#define CCH   64

typedef __attribute__((ext_vector_type(4))) int v4i;

// ---------------------------------------------------------------------------
// Kernel 1: zero the 256 MB canvas with 16B-wide stores.
// CDNA5 path: GLOBAL_STORE_ASYNC_FROM_LDS_B128 (ASYNCcnt-tracked) sourcing a
// zeroed LDS staging buffer — store data never touches VGPRs.
// ---------------------------------------------------------------------------
__global__ void zero_canvas_async(float* __restrict__ out,
                                  unsigned long long n4,   // number of float4 chunks
                                  unsigned tail)           // out_size % 4 leftover floats
{
    __shared__ __align__(16) float lds_zero[128];          // 512 B: 32 lanes x 16 B
    if (threadIdx.x < 128u) lds_zero[threadIdx.x] = 0.0f;
    __syncthreads();

    unsigned lane = threadIdx.x & 31u;
    unsigned long long i =
        (unsigned long long)blockIdx.x * blockDim.x + threadIdx.x;

    if (i < n4) {
        float* gp = out + i * 4ull;
#if __has_builtin(__builtin_amdgcn_global_store_async_from_lds_b128)
        typedef __attribute__((address_space(1))) v4i gv4i;
        typedef __attribute__((address_space(3))) v4i lv4i;
        __builtin_amdgcn_global_store_async_from_lds_b128(
            (gv4i*)(v4i*)gp,
            (lv4i*)(v4i*)&lds_zero[lane * 4u],
            /*offset=*/0, /*cpol=*/0);
#else
        asm volatile("global_store_async_from_lds_b128 %0, %1, off"
                     :: "v"(gp), "v"((unsigned)(uintptr_t)&lds_zero[lane * 4u])
                     : "memory");
#endif
    }

    // Tail (out_size not a multiple of 4) — not hit for this shape, safety only.
    if (tail && i == 0ull) {
        for (unsigned t = 0; t < tail; ++t) out[n4 * 4ull + t] = 0.0f;
    }

#if __has_builtin(__builtin_amdgcn_s_wait_asynccnt)
    __builtin_amdgcn_s_wait_asynccnt(0);
#else
    asm volatile("s_wait_asynccnt 0" ::: "memory");
#endif
}

// ---------------------------------------------------------------------------
// Kernel 2: scatter pillars. One wave32 per pillar:
//   lane L reads float2 {feat[m][2L], feat[m][2L+1]}  (256 B contiguous / wave)
//   and stores them to out[b][2L][y][x], out[b][2L+1][y][x] (1 MB stride).
// m is wave-uniform -> readfirstlane turns coordinate reads into scalar loads.
// ---------------------------------------------------------------------------
__global__ void scatter_pillars(const float* __restrict__ feat,
                                const int*   __restrict__ coords,
                                float*       __restrict__ out,
                                int M)
{
    unsigned gtid = blockIdx.x * blockDim.x + threadIdx.x;
    int m = __builtin_amdgcn_readfirstlane((int)(gtid >> 5));   // wave-uniform pillar id
    if (m >= M) return;

    unsigned lane = threadIdx.x & 31u;

    // coordinates row: [b, z(=0), y, x], int32
    int b = coords[4 * m + 0];
    int y = coords[4 * m + 2];
    int x = coords[4 * m + 3];

    unsigned long long base =
        (unsigned long long)b * ((unsigned long long)CCH * NYX) +
        (unsigned long long)y * NXD + (unsigned)x;

    const float2 v = *(const float2*)(feat + (unsigned long long)m * CCH + lane * 2u);

    out[base + (unsigned long long)(2u * lane)      * NYX] = v.x;
    out[base + (unsigned long long)(2u * lane + 1u) * NYX] = v.y;
}

// ---------------------------------------------------------------------------
// Launch
// ---------------------------------------------------------------------------
extern "C" void kernel_launch(void* const* d_in, const int* in_sizes, int n_in,
                              void* d_out, int out_size, void* d_ws, size_t ws_size,
                              hipStream_t stream)
{
    const float* feat   = (const float*)d_in[0];   // (M, 64) f32
    const int*   coords = (const int*)d_in[1];     // (M, 4)  i32
    float*       out    = (float*)d_out;           // (4, 64, 512, 512) f32

    const int M = in_sizes[0] / CCH;

    // Pass 1: zero canvas (256 MB), one B128 async store per thread.
    unsigned long long n4 = (unsigned long long)out_size / 4ull;
    unsigned tail = (unsigned)((unsigned long long)out_size & 3ull);
    unsigned zblocks = (unsigned)((n4 + 255ull) / 256ull);
    if (zblocks == 0) zblocks = 1;
    zero_canvas_async<<<zblocks, 256, 0, stream>>>(out, n4, tail);

    // Pass 2: scatter, one wave32 per pillar, 8 waves per 256-thread block.
    int sblocks = (M + 7) / 8;
    if (sblocks > 0) {
        scatter_pillars<<<sblocks, 256, 0, stream>>>(feat, coords, out, M);
    }
}